// MoGNN_26036091748364
// MI455X (gfx1250) — compile-verified
//
#include <hip/hip_runtime.h>
#include <hip/hip_bf16.h>

#define NUM_GRAPHS 1024
#define F_DIM      512
#define OUT_F      7

typedef __attribute__((ext_vector_type(2))) float v2f;
typedef __attribute__((ext_vector_type(8))) float v8f;

// ---------------------------------------------------------------------------
// Kernel 1: deterministic segment-mean pooling (batch ids are sorted).
// Block g binary-searches its node range [start,end) and streams x with
// float4 (b128) loads; 128 threads cover the 512-feature row. No atomics.
// ---------------------------------------------------------------------------
__global__ __launch_bounds__(128)
void segment_mean_kernel(const float* __restrict__ x,
                         const int*   __restrict__ batch,
                         float*       __restrict__ pooled,   // [NUM_GRAPHS, F_DIM]
                         int N) {
    const int g = blockIdx.x;           // graph id, 0..1023
    const int t = threadIdx.x;          // 0..127
    const int f0 = t * 4;               // feature base for this thread

    // lower_bound(batch, g)  -- uniform across block (scalar branches)
    int lo = 0, hi = N;
    while (lo < hi) { int mid = (lo + hi) >> 1; if (batch[mid] <  g) lo = mid + 1; else hi = mid; }
    const int start = lo;
    // upper_bound(batch, g)
    hi = N;
    while (lo < hi) { int mid = (lo + hi) >> 1; if (batch[mid] <= g) lo = mid + 1; else hi = mid; }
    const int end = lo;

    float ax = 0.f, ay = 0.f, az = 0.f, aw = 0.f;
    for (int n = start; n < end; ++n) {
        const float4 v = *reinterpret_cast<const float4*>(x + (size_t)n * F_DIM + f0);
        ax += v.x; ay += v.y; az += v.z; aw += v.w;
    }
    const float inv = 1.0f / fmaxf((float)(end - start), 1.0f);
    float4 r; r.x = ax * inv; r.y = ay * inv; r.z = az * inv; r.w = aw * inv;
    *reinterpret_cast<float4*>(pooled + (size_t)g * F_DIM + f0) = r;
}

// ---------------------------------------------------------------------------
// Kernel 2: out[1024,7] = pooled[1024,512] @ W[7,512]^T + b, via
// V_WMMA_F32_16X16X4_F32. One wave per 16-row tile (64 tiles), N padded to 16.
// A layout: lane = row + 16*half; VGPR j holds K = k0 + 2*half + j.
// B layout: lane = col + 16*half; same K mapping; B[k][n] = W[n][k] (n<7).
// D layout: VGPR r -> M = r + 8*half, N = lane&15.
//
// Padding lanes (row >= 7) do an unconditional load from a clamped W row and
// zero it with a 0/1 multiply -- no EXEC-mask branches in the WMMA loop.
// ---------------------------------------------------------------------------
__global__ __launch_bounds__(32)
void pool_gemm_wmma_kernel(const float* __restrict__ pooled,  // [1024, 512]
                           const float* __restrict__ W,       // [7, 512]
                           const float* __restrict__ b,       // [7]
                           float*       __restrict__ out) {   // [1024, 7]
    const int tile = blockIdx.x;            // 0..63
    const int lane = threadIdx.x & 31;
    const int row  = lane & 15;             // M for A, N for B/D
    const int half = lane >> 4;             // 0 or 1
    const int kofs = 2 * half;

    const float* __restrict__ arow = pooled + (size_t)(tile * 16 + row) * F_DIM;
    const int    wr     = (row < OUT_F) ? row : 0;      // clamped: always valid
    const float  bscale = (row < OUT_F) ? 1.0f : 0.0f;  // zero padding lanes
    const float* __restrict__ wrow = W + (size_t)wr * F_DIM;

    v8f c = {};   // f32 accumulator tile, starts at 0
    #pragma unroll 8
    for (int k0 = 0; k0 < F_DIM; k0 += 4) {
        const int k = k0 + kofs;            // even -> 8B aligned b64 loads
        const float2 av = *reinterpret_cast<const float2*>(arow + k);
        const float2 wv = *reinterpret_cast<const float2*>(wrow + k);
        v2f a, bm;
        a.x  = av.x;            a.y  = av.y;
        bm.x = wv.x * bscale;   bm.y = wv.y * bscale;
        // (neg_a, A, neg_b, B, c_mod, C, reuse_a, reuse_b)
        c = __builtin_amdgcn_wmma_f32_16x16x4_f32(false, a, false, bm,
                                                  (short)0, c, false, false);
    }

    // Store: D VGPR r -> M = tile*16 + 8*half + r, N = row; keep N < 7.
    if (row < OUT_F) {
        const float bias  = b[row];
        const int   mbase = tile * 16 + 8 * half;
        #pragma unroll
        for (int r = 0; r < 8; ++r) {
            out[(size_t)(mbase + r) * OUT_F + row] = c[r] + bias;
        }
    }
}

// ---------------------------------------------------------------------------
// Launch. Inputs (setup_inputs order): 0=x, 1=edge_index (unused),
// 2=edge_attr (unused), 3=batch_size, 4=W, 5=b. Output: [1024,7] f32.
// Workspace: pooled matrix, 1024*512 f32 = 2 MB.
// ---------------------------------------------------------------------------
extern "C" void kernel_launch(void* const* d_in, const int* in_sizes, int n_in,
                              void* d_out, int out_size, void* d_ws, size_t ws_size,
                              hipStream_t stream) {
    const float* x     = (const float*)d_in[0];
    const int*   batch = (const int*)  d_in[3];
    const float* W     = (const float*)d_in[4];
    const float* b     = (const float*)d_in[5];
    float*       out   = (float*)d_out;

    const int N = in_sizes[3];                 // number of nodes (100000)
    float* pooled = (float*)d_ws;              // [1024, 512]

    segment_mean_kernel<<<NUM_GRAPHS, 128, 0, stream>>>(x, batch, pooled, N);
    pool_gemm_wmma_kernel<<<NUM_GRAPHS / 16, 32, 0, stream>>>(pooled, W, b, out);
}